// EventWarping_57690000720628
// MI455X (gfx1250) — compile-verified
//
#include <hip/hip_runtime.h>
#include <hip/hip_bf16.h>

// ---------------- problem constants ----------------
constexpr int   B  = 8;
constexpr int   N  = 131072;
constexpr int   H  = 480;
constexpr int   W  = 640;
constexpr int   HW = H * W;
constexpr float FLOW_SCALING = 640.0f;   // max(H,W)
constexpr float REGUL_WEIGHT = 0.001f;
constexpr float EPS = 1e-9f;

constexpr float INV_NDX = 1.0f / (float)(B * H * (W - 1));       // dx mean denom
constexpr float INV_NDY = 1.0f / (float)(B * (H - 1) * W);       // dy mean denom
constexpr float INV_NDD = 1.0f / (float)(B * (H - 1) * (W - 1)); // dr/ur mean denom

typedef float v2f __attribute__((ext_vector_type(2)));
typedef float v8f __attribute__((ext_vector_type(8)));

// Wave(32)-wide sum using the CDNA5 matrix pipe:
// A(16x4 f32) holds one partial per lane (a[0]=x, a[1]=0); B = all-ones.
// D[m][n] = x_lane_m + x_lane_(m+16).  Lane 0 sums rows 0..7, lane 16 rows
// 8..15; shfl_xor(16) combines.  EXEC must be all-1s at the call site.
__device__ __forceinline__ float wave_sum_wmma(float x) {
  v2f a; a[0] = x;    a[1] = 0.0f;
  v2f b; b[0] = 1.0f; b[1] = 1.0f;
  v8f c = {};
  v8f d = __builtin_amdgcn_wmma_f32_16x16x4_f32(
      /*neg_a=*/false, a, /*neg_b=*/false, b,
      /*c_mod=*/(short)0, c, /*reuse_a=*/false, /*reuse_b=*/false);
  float s = d[0] + d[1] + d[2] + d[3] + d[4] + d[5] + d[6] + d[7];
  s += __shfl_xor(s, 16, 32);
  return s;  // every lane holds the full wave sum
}

// -------- pass 1: warp events and scatter-add bilinear weights --------
__global__ void ew_splat(const float* __restrict__ flow,
                         const float* __restrict__ ts,
                         const int*   __restrict__ ys,
                         const int*   __restrict__ xs,
                         const int*   __restrict__ pol,
                         float* __restrict__ imgs,   // [4][B][HW]: pos,neg,pos_ts,neg_ts
                         int isFw) {
  int gid = blockIdx.x * blockDim.x + threadIdx.x;
  if (gid >= B * N) return;
  int b = gid / N;

  int   y  = ys[gid];
  int   x  = xs[gid];
  float t0 = ts[gid];
  int   p  = pol[gid];

  int flat = y * W + x;
  const float* fb = flow + (size_t)b * 2 * HW;
  float fx = fb[flat];
  float fy = fb[HW + flat];

  float tref = isFw ? 1.0f : 0.0f;
  float dt   = tref - t0;
  float wy   = (float)y + dt * fy * FLOW_SCALING;
  float wx   = (float)x + dt * fx * FLOW_SCALING;
  float ty   = floorf(wy);
  float lx   = floorf(wx);
  float fyr  = wy - ty;           // in [0,1)
  float fxr  = wx - lx;
  float tsw  = isFw ? t0 : (1.0f - t0);

  float* img_w = imgs + (size_t)(p ? 0 : 1) * (B * HW) + (size_t)b * HW;
  float* img_t = img_w + (size_t)2 * (B * HW);

  float wyv[2] = {1.0f - fyr, fyr};
  float wxv[2] = {1.0f - fxr, fxr};
#pragma unroll
  for (int cy = 0; cy < 2; ++cy) {
    float iyf = ty + (float)cy;
    if (iyf < 0.0f || iyf >= (float)H) continue;
    int rowbase = (int)iyf * W;
#pragma unroll
    for (int cx = 0; cx < 2; ++cx) {
      float ixf = lx + (float)cx;
      if (ixf < 0.0f || ixf >= (float)W) continue;
      float wgt = wyv[cy] * wxv[cx];
      if (wgt == 0.0f) continue;
      int pix = rowbase + (int)ixf;
      atomicAdd(img_w + pix, wgt);
      atomicAdd(img_t + pix, wgt * tsw);
    }
  }
}

// -------- pass 2: per-batch loss + nonzero count (WMMA wave reduce) --------
__global__ void ew_reduce(const float* __restrict__ imgs,
                          float* __restrict__ partials) {  // [B][2] = {l, nz}
  int b = blockIdx.y;
  const float* pos  = imgs + (size_t)b * HW;
  const float* neg  = pos  + (size_t)1 * (B * HW);
  const float* post = pos  + (size_t)2 * (B * HW);
  const float* negt = pos  + (size_t)3 * (B * HW);

  float l = 0.0f, nz = 0.0f;
  for (int i = blockIdx.x * blockDim.x + threadIdx.x; i < HW;
       i += blockDim.x * gridDim.x) {
    float pw = pos[i], nw = neg[i], pt = post[i], nt = negt[i];
    float a = pt / (pw + EPS);      // MAX_TS == 1.0
    float c = nt / (nw + EPS);
    l  += a * a + c * c;
    nz += (pw + nw > 0.0f) ? 1.0f : 0.0f;
  }
  float lsum  = wave_sum_wmma(l);
  float nzsum = wave_sum_wmma(nz);
  if ((threadIdx.x & 31) == 0) {
    atomicAdd(&partials[2 * b + 0], lsum);
    atomicAdd(&partials[2 * b + 1], nzsum);
  }
}

__global__ void ew_finalize(const float* __restrict__ partials,
                            float* __restrict__ out) {
  if (blockIdx.x == 0 && threadIdx.x == 0) {
    float s = 0.0f;
#pragma unroll
    for (int b = 0; b < B; ++b) s += partials[2 * b] / partials[2 * b + 1];
    atomicAdd(out, s);
  }
}

// -------- pass 3: Charbonnier smoothness on the flow field --------
__global__ void ew_smooth(const float* __restrict__ flow,
                          float* __restrict__ out) {
  float acc = 0.0f;
  const int total = B * HW;
  for (int i = blockIdx.x * blockDim.x + threadIdx.x; i < total;
       i += blockDim.x * gridDim.x) {
    int b   = i / HW;
    int pix = i - b * HW;
    int y   = pix / W;
    int x   = pix - y * W;
    const float* fx = flow + (size_t)b * 2 * HW;
    const float* fy = fx + HW;
    float f0x = fx[pix], f0y = fy[pix];
    bool xe = (x < W - 1), ye = (y < H - 1);
    if (xe) {
      float dxx = f0x - fx[pix + 1], dxy = f0y - fy[pix + 1];
      acc += sqrtf(dxx * dxx + dxy * dxy + 1e-6f) * INV_NDX;
    }
    if (ye) {
      float dyx = f0x - fx[pix + W], dyy = f0y - fy[pix + W];
      acc += sqrtf(dyx * dyx + dyy * dyy + 1e-6f) * INV_NDY;
    }
    if (xe && ye) {
      float drx = f0x - fx[pix + W + 1], dry = f0y - fy[pix + W + 1];
      acc += sqrtf(drx * drx + dry * dry + 1e-6f) * INV_NDD;
      float urx = fx[pix + W] - fx[pix + 1], ury = fy[pix + W] - fy[pix + 1];
      acc += sqrtf(urx * urx + ury * ury + 1e-6f) * INV_NDD;
    }
  }
  float s = wave_sum_wmma(acc);
  if ((threadIdx.x & 31) == 0) atomicAdd(out, s * (REGUL_WEIGHT * 0.25f));
}

// ---------------- launcher ----------------
extern "C" void kernel_launch(void* const* d_in, const int* in_sizes, int n_in,
                              void* d_out, int out_size, void* d_ws, size_t ws_size,
                              hipStream_t stream) {
  (void)in_sizes; (void)n_in; (void)out_size; (void)ws_size;
  const float* flow = (const float*)d_in[0];
  const float* ts   = (const float*)d_in[1];
  const int*   ys   = (const int*)d_in[2];
  const int*   xs   = (const int*)d_in[3];
  const int*   pol  = (const int*)d_in[4];
  float* out = (float*)d_out;

  float* imgs     = (float*)d_ws;                       // 4*B*HW floats (~39.3 MB)
  float* partials = imgs + (size_t)4 * B * HW;          // B*2 floats
  size_t zbytes   = ((size_t)4 * B * HW + 2 * B) * sizeof(float);

  hipMemsetAsync(out, 0, sizeof(float), stream);

  const int splat_blocks = (B * N + 255) / 256;
  dim3 rgrid(300, B);  // 76800 threads per batch -> 4 pixels each

  for (int isFw = 1; isFw >= 0; --isFw) {
    hipMemsetAsync(imgs, 0, zbytes, stream);
    ew_splat<<<splat_blocks, 256, 0, stream>>>(flow, ts, ys, xs, pol, imgs, isFw);
    ew_reduce<<<rgrid, 256, 0, stream>>>(imgs, partials);
    ew_finalize<<<1, 1, 0, stream>>>(partials, out);
  }
  ew_smooth<<<2400, 256, 0, stream>>>(flow, out);
}